// MaskExtractor_47038481826290
// MI455X (gfx1250) — compile-verified
//
#include <hip/hip_runtime.h>
#include <hip/hip_bf16.h>
#include <math.h>

#define NUM_IMGS 16
#define MASKS    8
#define NG       (NUM_IMGS*MASKS)   // 128 mask groups
#define HP       576                // 24*24 patches
#define NP       24
#define CIN      1024
#define CHID     4096
#define HM       336
#define RTOK     4

typedef __attribute__((ext_vector_type(2))) float v2f;
typedef __attribute__((ext_vector_type(8))) float v8f;

// Async global->LDS path (CDNA5): guarded so compile can never regress.
#if defined(__gfx1250__) && __has_builtin(__builtin_amdgcn_global_load_async_to_lds_b128)
#define GEMM_ASYNC 1
#else
#define GEMM_ASYNC 0
#endif

#if GEMM_ASYNC
typedef __attribute__((ext_vector_type(4))) int v4i;
typedef __attribute__((address_space(1))) v4i* gptr_b128;   // global int4*
typedef __attribute__((address_space(3))) v4i* lptr_b128;   // LDS int4*
#if __has_builtin(__builtin_amdgcn_s_wait_asynccnt)
#define ASYNC_WAIT() __builtin_amdgcn_s_wait_asynccnt(0)
#else
#define ASYNC_WAIT() asm volatile("s_wait_asynccnt 0x0" ::: "memory")
#endif
#else
#define ASYNC_WAIT() do {} while (0)
#endif

// ---------------------------------------------------------------------------
// Kernel 1: bilinear(antialias triangle, radius 14) resize 336->24, >0 mask,
// masked average pool over 576 patch positions of [576,1024] features.
// One block per (img,mask) group; fully coalesced float4 feature streaming.
// ---------------------------------------------------------------------------
__global__ __launch_bounds__(256) void k_mask_pool(const float* __restrict__ feats,
                                                   const float* __restrict__ masks,
                                                   float* __restrict__ pooled) {
  __shared__ float tmp[NP * HM];   // 24 x 336 vertical-resized
  __shared__ float mres[HP];       // 24 x 24 binary mask
  __shared__ float red[256];
  const int g = blockIdx.x;
  const int tid = threadIdx.x;
  const float* mk = masks + (size_t)g * HM * HM;
  const float inv14 = 1.0f / 14.0f;

  // phase A: vertical resize 336 -> 24 (triangle kernel, support 28 taps)
  for (int idx = tid; idx < NP * HM; idx += 256) {
    const int oy = idx / HM, x = idx - oy * HM;
    const float c = 14.0f * oy + 6.5f;
    int i0 = 14 * oy - 7;  if (i0 < 0) i0 = 0;
    int i1 = 14 * oy + 20; if (i1 > HM - 1) i1 = HM - 1;
    float acc = 0.0f;
    for (int iy = i0; iy <= i1; ++iy) {
      const float w = 1.0f - fabsf((float)iy - c) * inv14;
      acc += w * mk[iy * HM + x];
    }
    tmp[idx] = acc;   // positive weight normalization is sign-preserving
  }
  __syncthreads();

  // phase B: horizontal resize + threshold > 0
  for (int idx = tid; idx < HP; idx += 256) {
    const int oy = idx / NP, ox = idx - oy * NP;
    const float c = 14.0f * ox + 6.5f;
    int i0 = 14 * ox - 7;  if (i0 < 0) i0 = 0;
    int i1 = 14 * ox + 20; if (i1 > HM - 1) i1 = HM - 1;
    float acc = 0.0f;
    for (int ix = i0; ix <= i1; ++ix) {
      const float w = 1.0f - fabsf((float)ix - c) * inv14;
      acc += w * tmp[oy * HM + ix];
    }
    mres[idx] = (acc > 0.0f) ? 1.0f : 0.0f;
  }
  __syncthreads();

  // count active cells (denorm)
  float cacc = 0.0f;
  for (int p = tid; p < HP; p += 256) cacc += mres[p];
  red[tid] = cacc;
  __syncthreads();
  for (int s = 128; s > 0; s >>= 1) {
    if (tid < s) red[tid] += red[tid + s];
    __syncthreads();
  }
  const float inv = 1.0f / (red[0] + 1e-8f);

  // masked average pool: each thread owns 4 contiguous channels
  const int c0 = tid * 4;
  float a0 = 0.f, a1 = 0.f, a2 = 0.f, a3 = 0.f;
  const float* fb = feats + (size_t)g * HP * CIN + c0;
  for (int p = 0; p < HP; ++p) {
    if (mres[p] > 0.0f) {                 // block-uniform branch
      const float4 f = *(const float4*)(fb + (size_t)p * CIN);
      a0 += f.x; a1 += f.y; a2 += f.z; a3 += f.w;
    }
  }
  float4 o; o.x = a0 * inv; o.y = a1 * inv; o.z = a2 * inv; o.w = a3 * inv;
  *(float4*)(pooled + (size_t)g * CIN + c0) = o;
}

// ---------------------------------------------------------------------------
// Kernel 2: token merge — 8 pooled tokens -> 4 region tokens per image.
// ---------------------------------------------------------------------------
__global__ __launch_bounds__(256) void k_token_merge(const float* __restrict__ pooled,
                                                     float* __restrict__ tokens) {
  __shared__ float pl[MASKS * CIN];      // 32 KB token tile
  __shared__ float stat[16];             // n2[0..7], dot[8..14]
  __shared__ int   segs[MASKS];
  __shared__ int   cnts[RTOK];
  const int img = blockIdx.x, tid = threadIdx.x;
  const float* src = pooled + (size_t)img * MASKS * CIN;

  for (int i = tid; i < (MASKS * CIN) / 4; i += 256)
    *(float4*)(pl + i * 4) = *(const float4*)(src + i * 4);
  __syncthreads();

  if (tid < 15) {
    float s = 0.0f;
    if (tid < 8) {
      const float* a = pl + tid * CIN;
      for (int c = 0; c < CIN; ++c) s += a[c] * a[c];
    } else {
      const float* a = pl + (tid - 8) * CIN;
      const float* b = a + CIN;
      for (int c = 0; c < CIN; ++c) s += a[c] * b[c];
    }
    stat[tid] = s;
  }
  __syncthreads();

  if (tid == 0) {
    float nrm[MASKS];
    for (int m = 0; m < MASKS; ++m) nrm[m] = fmaxf(sqrtf(stat[m]), 1e-12f);
    float sim[MASKS - 1], v[MASKS - 1];
    for (int i = 0; i < MASKS - 1; ++i) {
      sim[i] = stat[8 + i] / (nrm[i] * nrm[i + 1]);
      v[i] = sim[i];
    }
    // 4th largest of 7 (partial selection sort)
    for (int a = 0; a < RTOK; ++a) {
      int mi = a;
      for (int b = a + 1; b < MASKS - 1; ++b) if (v[b] > v[mi]) mi = b;
      const float t = v[a]; v[a] = v[mi]; v[mi] = t;
    }
    const float kth = v[RTOK - 1];
    int cnt[RTOK]; for (int r = 0; r < RTOK; ++r) cnt[r] = 0;
    int s = 0; segs[0] = 0; cnt[0] = 1;
    for (int m = 1; m < MASKS; ++m) {
      if (sim[m - 1] < kth) ++s;           // strict cut => s <= 3 always
      segs[m] = s; cnt[s]++;
    }
    for (int r = 0; r < RTOK; ++r) cnts[r] = cnt[r];
  }
  __syncthreads();

  for (int c = tid; c < CIN; c += 256) {
    float acc[RTOK] = {0.f, 0.f, 0.f, 0.f};
    for (int m = 0; m < MASKS; ++m) acc[segs[m]] += pl[m * CIN + c];
    for (int r = 0; r < RTOK; ++r)
      tokens[((size_t)img * RTOK + r) * CIN + c] = acc[r] / (float)max(cnts[r], 1);
  }
}

// ---------------------------------------------------------------------------
// Kernels 3/4: C[64,N] = act(A[64,K] @ W[K,N] + b) with f32 WMMA 16x16x4.
// Block = 8 waves; wave -> one 16-col tile, 2 M-tiles. A chunks (64 x 128,
// +4 float pad) double-buffered in LDS via GLOBAL_LOAD_ASYNC_TO_LDS_B128
// (ASYNCcnt) so the next chunk's fetch overlaps the current WMMA inner loop.
// W streams from global (L2-resident on 192MB L2), register-reused across
// both M-tiles.
// ---------------------------------------------------------------------------
#define KC  128
#define LDP (KC + 4)   // row stride 528B: kills 512B-stride bank conflicts, 16B aligned

__global__ __launch_bounds__(256) void k_gemm64(const float* __restrict__ A,
                                                const float* __restrict__ W,
                                                const float* __restrict__ bias,
                                                float* __restrict__ out,
                                                int K, int N, int gelu) {
  __shared__ float As[2][64 * LDP];   // ~66 KB double buffer
  const int tid  = threadIdx.x;
  const int wave = tid >> 5;          // wave32
  const int lane = tid & 31;
  const int half = lane >> 4;         // 0: K=k,k+1 | 1: K=k+2,k+3
  const int col  = lane & 15;
  const int mgrp = wave & 1;                       // rows mgrp*32 .. +31
  const int n0   = (blockIdx.x * 4 + (wave >> 1)) * 16;

  v8f acc0 = {}; v8f acc1 = {};
  const float bval = bias[n0 + col];
  const int nchunks = K / KC;

  // stage A[0:64, kc:kc+KC) into LDS buffer `buf`
  auto stage = [&](int buf, int kc) {
    for (int i = tid; i < 64 * (KC / 4); i += 256) {
      const int r = i / (KC / 4), cv = i - r * (KC / 4);
#if GEMM_ASYNC
      __builtin_amdgcn_global_load_async_to_lds_b128(
          (gptr_b128)(const_cast<float*>(A) + (size_t)r * K + kc + cv * 4),
          (lptr_b128)&As[buf][r * LDP + cv * 4], 0, 0);
#else
      *(float4*)&As[buf][r * LDP + cv * 4] =
          *(const float4*)(A + (size_t)r * K + kc + cv * 4);
#endif
    }
  };

  stage(0, 0);
  for (int ch = 0; ch < nchunks; ++ch) {
    ASYNC_WAIT();        // my async transfers for buffer ch&1 are in LDS
    __syncthreads();     // everyone's are; prior reads of the other buffer done
    if (ch + 1 < nchunks) stage((ch + 1) & 1, (ch + 1) * KC);

    const float* Ac = As[ch & 1];
    const float* Wp = W + (size_t)(ch * KC) * N + n0 + col;
#pragma unroll 4
    for (int k = 0; k < KC; k += 4) {
      const int kk = k + 2 * half;
      const float* wrow = Wp + (size_t)kk * N;
      v2f b; b.x = wrow[0]; b.y = wrow[(size_t)N];          // B[kk,n], B[kk+1,n]
      v2f a0 = *(const v2f*)&Ac[(mgrp * 32 +      col) * LDP + kk];
      v2f a1 = *(const v2f*)&Ac[(mgrp * 32 + 16 + col) * LDP + kk];
      acc0 = __builtin_amdgcn_wmma_f32_16x16x4_f32(false, a0, false, b,
                                                   (short)0, acc0, false, false);
      acc1 = __builtin_amdgcn_wmma_f32_16x16x4_f32(false, a1, false, b,
                                                   (short)0, acc1, false, false);
    }
  }

  // epilogue: C/D layout — VGPR j: lanes0-15 M=j, lanes16-31 M=8+j, N=lane&15
  for (int j = 0; j < 8; ++j) {
    const int r0 = mgrp * 32 + j + 8 * half;
    const int r1 = r0 + 16;
    float x0 = acc0[j] + bval;
    float x1 = acc1[j] + bval;
    if (gelu) {
      x0 = 0.5f * x0 * (1.0f + erff(x0 * 0.7071067811865475f));
      x1 = 0.5f * x1 * (1.0f + erff(x1 * 0.7071067811865475f));
    }
    out[(size_t)r0 * N + n0 + col] = x0;
    out[(size_t)r1 * N + n0 + col] = x1;
  }
}

// ---------------------------------------------------------------------------
extern "C" void kernel_launch(void* const* d_in, const int* in_sizes, int n_in,
                              void* d_out, int out_size, void* d_ws, size_t ws_size,
                              hipStream_t stream) {
  const float* feats = (const float*)d_in[0];   // [128, 576, 1024]
  const float* masks = (const float*)d_in[1];   // [16, 8, 336, 336]
  const float* W1    = (const float*)d_in[2];   // [1024, 4096]
  const float* b1    = (const float*)d_in[3];   // [4096]
  const float* W2    = (const float*)d_in[4];   // [4096, 4096]
  const float* b2    = (const float*)d_in[5];   // [4096]
  float* outp = (float*)d_out;                  // [64, 4096]

  float* pooled = (float*)d_ws;                   // 128*1024 f32
  float* tokens = pooled + NG * CIN;              // 64*1024 f32
  float* hbuf   = tokens + NUM_IMGS * RTOK * CIN; // 64*4096 f32

  k_mask_pool  <<<NG,        256, 0, stream>>>(feats, masks, pooled);
  k_token_merge<<<NUM_IMGS,  256, 0, stream>>>(pooled, tokens);
  k_gemm64     <<<CHID / 64, 256, 0, stream>>>(tokens, W1, b1, hbuf, CIN,  CHID, 1);
  k_gemm64     <<<CHID / 64, 256, 0, stream>>>(hbuf,   W2, b2, outp, CHID, CHID, 0);
}